// HyperbolicInfoNCE_82394652606599
// MI455X (gfx1250) — compile-verified
//
#include <hip/hip_runtime.h>

// ---------------- problem constants ----------------
#define BN      8192          // batch (rows == cols of sims)
#define D1      129           // ambient Lorentz dim
#define KP      132           // padded K (multiple of 4)
#define NEG_INV_TEMP (-14.285714285714285714f)   // -1/0.07
#define ONE_EPS 1.000001f     // 1 + 1e-6

typedef __attribute__((ext_vector_type(2))) float v2f;
typedef __attribute__((ext_vector_type(8))) float v8f;

// ---------------- kernel 1: pad + fold Lorentz metric ----------------
__global__ void hyp_prep(const float* __restrict__ z1,
                         const float* __restrict__ z2,
                         float* __restrict__ z1p,
                         float* __restrict__ z2p) {
    int idx = blockIdx.x * blockDim.x + threadIdx.x;
    if (idx >= BN * KP) return;
    int r = idx / KP;
    int c = idx - r * KP;
    float v1 = 0.0f, v2 = 0.0f;
    if (c < D1) {
        v1 = z1[r * D1 + c];
        v2 = z2[r * D1 + c];
        if (c == 0) v2 = -v2;     // metric: negate time-like coordinate
    }
    z1p[idx] = v1;
    z2p[idx] = v2;
}

// ---------------- kernel 2: fused WMMA GEMM + acosh/exp + reductions ----------------
// grid (64,64); block 256 threads = 8 waves; wave tile 32x64 (2x4 WMMA 16x16 tiles)
__global__ __launch_bounds__(256, 2)
void hyp_gemm(const float* __restrict__ z1p,
              const float* __restrict__ z2p,
              float* __restrict__ rowsum,
              float* __restrict__ colsum,
              float* __restrict__ diagsum) {
    const int wave = threadIdx.x >> 5;
    const int lane = threadIdx.x & 31;
    const int l15  = lane & 15;
    const int kh   = lane >> 4;          // 0 or 1: which K half of the 4-chunk

    const int bm = blockIdx.x;
    const int bn = blockIdx.y;
    const int M0 = bm * 128 + (wave & 3) * 32;   // wave's first row
    const int N0 = bn * 128 + (wave >> 2) * 64;  // wave's first col

    v8f acc[2][4] = {};   // 2 M-tiles x 4 N-tiles of 16x16 f32 accumulators

    // Fragment base pointers: lane l<16 holds row l, K = k0 + 2*kh + {0,1}
    const float* aB = z1p + (size_t)(M0 + l15) * KP + kh * 2;
    const float* bB = z2p + (size_t)(N0 + l15) * KP + kh * 2;

    for (int k0 = 0; k0 < KP; k0 += 4) {
        v2f a[2], b[4];
#pragma unroll
        for (int mt = 0; mt < 2; ++mt)
            a[mt] = *(const v2f*)(aB + (size_t)mt * 16 * KP + k0);
#pragma unroll
        for (int nt = 0; nt < 4; ++nt)
            b[nt] = *(const v2f*)(bB + (size_t)nt * 16 * KP + k0);
#pragma unroll
        for (int mt = 0; mt < 2; ++mt)
#pragma unroll
            for (int nt = 0; nt < 4; ++nt)
                acc[mt][nt] = __builtin_amdgcn_wmma_f32_16x16x4_f32(
                    false, a[mt], false, b[nt], (short)0, acc[mt][nt],
                    false, false);
    }

    // ---- fused epilogue: sims = -acosh(max(-inner, 1+eps))/TEMP ; e = exp(sims)
    const bool diagBlock = (bm == bn);
    float rowPart[2][8];
#pragma unroll
    for (int mt = 0; mt < 2; ++mt)
#pragma unroll
        for (int v = 0; v < 8; ++v) rowPart[mt][v] = 0.0f;
    float diagAcc = 0.0f;

#pragma unroll
    for (int mt = 0; mt < 2; ++mt) {
#pragma unroll
        for (int nt = 0; nt < 4; ++nt) {
            float colPart = 0.0f;
#pragma unroll
            for (int v = 0; v < 8; ++v) {
                float inner = acc[mt][nt][v];
                float x    = fmaxf(-inner, ONE_EPS);
                float dist = __logf(x + __fsqrt_rn(x * x - 1.0f));  // acosh
                float sims = dist * NEG_INV_TEMP;
                float e    = __expf(sims);
                if (diagBlock) {
                    int R = M0 + mt * 16 + kh * 8 + v;   // C/D layout: M = v + 8*kh
                    int C = N0 + nt * 16 + l15;          //             N = lane&15
                    if (R == C) diagAcc += sims;
                }
                rowPart[mt][v] += e;
                colPart        += e;
            }
            // column sums: fold the two 16-lane halves, one atomic per column
            colPart += __shfl_xor(colPart, 16, 32);
            if (kh == 0)
                atomicAdd(&colsum[N0 + nt * 16 + l15], colPart);
        }
    }

    // row sums: butterfly over the 16 lanes of each half (N within the tile)
#pragma unroll
    for (int mt = 0; mt < 2; ++mt) {
#pragma unroll
        for (int v = 0; v < 8; ++v) {
            float r = rowPart[mt][v];
            r += __shfl_xor(r, 1, 32);
            r += __shfl_xor(r, 2, 32);
            r += __shfl_xor(r, 4, 32);
            r += __shfl_xor(r, 8, 32);
            if (l15 == 0)
                atomicAdd(&rowsum[M0 + mt * 16 + kh * 8 + v], r);
        }
    }

    if (diagBlock) {
        diagAcc += __shfl_xor(diagAcc, 1, 32);
        diagAcc += __shfl_xor(diagAcc, 2, 32);
        diagAcc += __shfl_xor(diagAcc, 4, 32);
        diagAcc += __shfl_xor(diagAcc, 8, 32);
        diagAcc += __shfl_xor(diagAcc, 16, 32);
        if (lane == 0) atomicAdd(diagsum, diagAcc);
    }
}

// ---------------- kernel 3: final scalar reduction ----------------
__global__ void hyp_finalize(const float* __restrict__ rowsum,
                             const float* __restrict__ colsum,
                             const float* __restrict__ diagsum,
                             float* __restrict__ out) {
    __shared__ float red[256];
    float local = 0.0f;
    for (int i = threadIdx.x; i < BN; i += 256)
        local += __logf(rowsum[i]) + __logf(colsum[i]);
    red[threadIdx.x] = local;
    __syncthreads();
    for (int s = 128; s > 0; s >>= 1) {
        if ((int)threadIdx.x < s) red[threadIdx.x] += red[threadIdx.x + s];
        __syncthreads();
    }
    if (threadIdx.x == 0) {
        // loss = 0.5*mean(logsumexp_row) + 0.5*mean(logsumexp_col) - mean(diag)
        out[0] = red[0] * (0.5f / (float)BN) - diagsum[0] * (1.0f / (float)BN);
    }
}

// ---------------- host launcher ----------------
extern "C" void kernel_launch(void* const* d_in, const int* in_sizes, int n_in,
                              void* d_out, int out_size, void* d_ws, size_t ws_size,
                              hipStream_t stream) {
    const float* z1 = (const float*)d_in[0];
    const float* z2 = (const float*)d_in[1];

    float* z1p    = (float*)d_ws;                   // BN*KP
    float* z2p    = z1p + (size_t)BN * KP;          // BN*KP
    float* rowsum = z2p + (size_t)BN * KP;          // BN
    float* colsum = rowsum + BN;                    // BN
    float* diagsum = colsum + BN;                   // 1 (+pad)

    // zero the reduction buffers (rowsum, colsum, diagsum)
    hipMemsetAsync(rowsum, 0, (size_t)(2 * BN + 4) * sizeof(float), stream);

    int prepN = BN * KP;
    hyp_prep<<<(prepN + 255) / 256, 256, 0, stream>>>(z1, z2, z1p, z2p);

    dim3 grid(BN / 128, BN / 128, 1);   // 64 x 64 tiles of 128x128
    hyp_gemm<<<grid, 256, 0, stream>>>(z1p, z2p, rowsum, colsum, diagsum);

    hyp_finalize<<<1, 256, 0, stream>>>(rowsum, colsum, diagsum, (float*)d_out);
}